// ForgetMultCPU_30382598652077
// MI455X (gfx1250) — compile-verified
//
#include <hip/hip_runtime.h>
#include <stdint.h>

// Problem shape (fixed by the reference): x,f : [B,T,H] f32, out : [B,T,H] f32
#define B_ 32
#define T_ 2048
#define H_ 1024

#define CH      128            // channels per workgroup
#define NBLK_H  (H_ / CH)      // 8
#define TILE_T  32             // time steps staged per LDS tile
#define DEPTH   4              // quad buffering (3 tiles prefetched ahead)
#define NTILES  (T_ / TILE_T)  // 64
#define IPT     16             // async b128 instructions per wave per tile
                               // (64 rows/tile of 512B, 4 waves -> 16 each)

__global__ __launch_bounds__(CH) void forget_mult_scan(
    const float* __restrict__ x, const float* __restrict__ f,
    const float* __restrict__ h_init, float* __restrict__ out) {
  // [buffer][t][channel], channel contiguous -> conflict-free ds reads.
  // 4 * 32 * 128 * 2 * 4B = 128 KB of the 320 KB WGP LDS.
  __shared__ __align__(16) float xs[DEPTH][TILE_T][CH];
  __shared__ __align__(16) float fs[DEPTH][TILE_T][CH];

  const int tid  = threadIdx.x;
  const int wave = tid >> 5;
  const int lane = tid & 31;
  const int b    = blockIdx.y;       // batch
  const int h0   = blockIdx.x * CH;  // channel-tile origin

  const size_t base = (size_t)b * T_ * H_ + h0;
  const float* xb = x + base;
  const float* fb = f + base;
  float*       ob = out + base;

  // Cooperatively issue async DMA of tile `ti` into LDS buffer ti % DEPTH.
  // Row (a, t): 128 contiguous floats = 512B = one b128 async op per wave
  // (each lane moves 16B). Wave w handles rows [w*IPT, w*IPT+IPT).
  auto issue_tile = [&](int ti) {
    const int buf = ti % DEPTH;
    const int t0  = ti * TILE_T;
#pragma unroll
    for (int k = 0; k < IPT; ++k) {
      const int p = wave * IPT + k;     // 0..63
      const int a = p >> 5;             // 0 = x, 1 = f
      const int t = p & (TILE_T - 1);
      const float* g = (a == 0 ? xb : fb) + (size_t)(t0 + t) * H_ + lane * 4;
      float* l = &(a == 0 ? xs : fs)[buf][t][lane * 4];
      const uint32_t lds_off = (uint32_t)(uintptr_t)l;  // low 32b of flat = LDS offset
      const uint64_t gaddr   = (uint64_t)(uintptr_t)g;
      asm volatile("global_load_async_to_lds_b128 %0, %1, off"
                   :
                   : "v"(lds_off), "v"(gaddr)
                   : "memory");
    }
  };

  // Prologue: prefetch DEPTH-1 tiles (48 async ops/wave outstanding).
  issue_tile(0);
  issue_tile(1);
  issue_tile(2);

  float h = h_init[(size_t)b * H_ + h0 + tid];

  for (int ti = 0; ti < NTILES; ++ti) {
    // Async loads retire in order: waiting ASYNCcnt <= 2*IPT guarantees this
    // wave's tile-ti ops are done while tiles ti+1, ti+2 stay in flight.
    if (ti < NTILES - 2)
      asm volatile("s_wait_asynccnt 32" ::: "memory");
    else if (ti == NTILES - 2)
      asm volatile("s_wait_asynccnt 16" ::: "memory");
    else
      asm volatile("s_wait_asynccnt 0" ::: "memory");
    __syncthreads();  // other waves' DMA for this tile also landed

    // Refill: tile ti+3 targets buffer (ti-1)%DEPTH, whose readers all
    // passed the trailing barrier of iteration ti-1. Outstanding stays
    // <= 3*IPT = 48 < 63 (ASYNCcnt cap), so issue never stalls.
    if (ti + DEPTH - 1 < NTILES) issue_tile(ti + DEPTH - 1);

    const int buf = ti % DEPTH;
    const int t0  = ti * TILE_T;
#pragma unroll
    for (int t = 0; t < TILE_T; ++t) {
      const float xv = xs[buf][t][tid];
      const float fv = fs[buf][t][tid];
      // h = f*x + (1-f)*h  ==  h + f*(x - h)
      h = __builtin_fmaf(fv, xv - h, h);
      __builtin_nontemporal_store(h, ob + (size_t)(t0 + t) * H_ + tid);
    }
    __syncthreads();  // everyone done reading buf before it is refilled
  }
}

extern "C" void kernel_launch(void* const* d_in, const int* in_sizes, int n_in,
                              void* d_out, int out_size, void* d_ws, size_t ws_size,
                              hipStream_t stream) {
  (void)in_sizes; (void)n_in; (void)out_size; (void)d_ws; (void)ws_size;
  const float* x  = (const float*)d_in[0];
  const float* f  = (const float*)d_in[1];
  const float* h0 = (const float*)d_in[2];
  float* out = (float*)d_out;

  dim3 grid(NBLK_H, B_);  // (8, 32) = 256 workgroups
  dim3 block(CH);         // 128 threads = 4 wave32
  forget_mult_scan<<<grid, block, 0, stream>>>(x, f, h0, out);
}